// MergeOutput_58858231824722
// MI455X (gfx1250) — compile-verified
//
#include <hip/hip_runtime.h>
#include <hip/hip_bf16.h>
#include <stdint.h>

// Soft-NMS on MI455X (gfx1250): one workgroup per batch, all state in LDS.
// Latency-bound sequential scan; CDNA5 paths used: async global->LDS staging
// (ASYNCcnt), wave32 shuffle argmax reduction, split-barrier workgroup sync.

#define NBOX    2048
#define NBATCH  16
#define THREADS 512
#define PER     (NBOX / THREADS)   // 4 boxes per thread
#define NW      (THREADS / 32)     // 16 waves
#define THRESH  0.3f
#define SIGMA   0.5f

__device__ __forceinline__ void async_load_b32_to_lds(uint32_t lds_off, const float* gsrc) {
    uint64_t ga = (uint64_t)(uintptr_t)gsrc;
    // vdst = LDS byte address (per-lane), vaddr = 64-bit global address, no saddr
    asm volatile("global_load_async_to_lds_b32 %0, %1, off"
                 :: "v"(lds_off), "v"(ga) : "memory");
}

__device__ __forceinline__ void wait_async0() {
#if __has_builtin(__builtin_amdgcn_s_wait_asynccnt)
    __builtin_amdgcn_s_wait_asynccnt(0);
#else
    asm volatile("s_wait_asynccnt 0" ::: "memory");
#endif
}

__global__ __launch_bounds__(THREADS)
void soft_nms_kernel(const float* __restrict__ det_g,
                     float* __restrict__ out_final,
                     float* __restrict__ out_keep) {
    // det[j*5 + {0..3}] = x1,y1,x2,y2 ; slot 4 holds score at load, area after init
    __shared__ float det[NBOX * 5];       // 40 KB
    __shared__ float key[NBOX];           // active ? current score : -1
    __shared__ float fin[NBOX];           // score at pick time, else -1
    __shared__ float redK[NW];
    __shared__ int   redI[NW];

    const int tid = threadIdx.x;
    const int b   = blockIdx.x;
    const float* src = det_g + (size_t)b * NBOX * 5;

    // ---- Stage this batch's detections into LDS via async copies (ASYNCcnt) ----
    #pragma unroll
    for (int k = 0; k < (NBOX * 5) / THREADS; ++k) {   // 20 dwords per thread
        int m = tid + k * THREADS;
        async_load_b32_to_lds((uint32_t)(uintptr_t)&det[m], src + m);
    }
    wait_async0();      // each wave waits for its own async copies
    __syncthreads();    // ... then all waves' LDS writes are visible

    // ---- Init: key/fin for owned elements; overwrite score slot with area ----
    #pragma unroll
    for (int k = 0; k < PER; ++k) {
        int j = tid + k * THREADS;
        float x1 = det[j * 5 + 0], y1 = det[j * 5 + 1];
        float x2 = det[j * 5 + 2], y2 = det[j * 5 + 3];
        float s  = det[j * 5 + 4];
        det[j * 5 + 4] = (x2 - x1 + 1.0f) * (y2 - y1 + 1.0f);   // area
        key[j] = (s > THRESH) ? s : -1.0f;
        fin[j] = -1.0f;
    }
    __syncthreads();

    // ---- Sequential soft-NMS scan ----
    for (int t = 0; t < NBOX; ++t) {
        // Thread-local argmax over owned boxes (ascending j + strict '>' keeps lowest index)
        float bk = -2.0f; int bi = 0;
        #pragma unroll
        for (int k = 0; k < PER; ++k) {
            int j = tid + k * THREADS;
            float v = key[j];
            if (v > bk) { bk = v; bi = j; }
        }
        // wave32 butterfly reduction, tie-break to lowest index (matches jnp.argmax)
        #pragma unroll
        for (int off = 16; off > 0; off >>= 1) {
            float ok = __shfl_xor(bk, off, 32);
            int   oi = __shfl_xor(bi, off, 32);
            if (ok > bk || (ok == bk && oi < bi)) { bk = ok; bi = oi; }
        }
        if ((tid & 31) == 0) { redK[tid >> 5] = bk; redI[tid >> 5] = bi; }
        __syncthreads();

        // All threads redundantly reduce the 16 wave results (identical answer everywhere)
        float bk2 = -2.0f; int bi2 = 0;
        #pragma unroll
        for (int w = 0; w < NW; ++w) {
            float v = redK[w]; int i = redI[w];
            if (v > bk2 || (v == bk2 && i < bi2)) { bk2 = v; bi2 = i; }
        }
        if (bk2 < 0.0f) break;   // no active boxes left; uniform across the block

        if ((bi2 & (THREADS - 1)) == tid) {   // owner of picked box
            key[bi2] = -1.0f;                 // deactivate
            fin[bi2] = bk2;                   // record score at pick time
        }

        // Broadcast read of the picked box (same LDS address for all lanes)
        float bx1 = det[bi2 * 5 + 0], by1 = det[bi2 * 5 + 1];
        float bx2 = det[bi2 * 5 + 2], by2 = det[bi2 * 5 + 3];
        float bar = det[bi2 * 5 + 4];

        // Decay active scores by exp(-iou^2 / sigma)
        #pragma unroll
        for (int k = 0; k < PER; ++k) {
            int j = tid + k * THREADS;
            float kj = key[j];
            if (kj >= 0.0f) {
                float xx1 = fmaxf(bx1, det[j * 5 + 0]);
                float yy1 = fmaxf(by1, det[j * 5 + 1]);
                float xx2 = fminf(bx2, det[j * 5 + 2]);
                float yy2 = fminf(by2, det[j * 5 + 3]);
                float w = fmaxf(xx2 - xx1 + 1.0f, 0.0f);
                float h = fmaxf(yy2 - yy1 + 1.0f, 0.0f);
                float inter = w * h;
                float iou = inter / (bar + det[j * 5 + 4] - inter);
                key[j] = kj * __expf(-iou * iou * (1.0f / SIGMA));
            }
        }
        __syncthreads();   // protects redK/redI reuse next iteration
    }

    // ---- Epilogue: final + keep (owner-private, no barrier needed) ----
    #pragma unroll
    for (int k = 0; k < PER; ++k) {
        int j = tid + k * THREADS;
        float f = fin[j];
        out_final[(size_t)b * NBOX + j] = f;
        out_keep [(size_t)b * NBOX + j] = (f > THRESH) ? 1.0f : 0.0f;
    }
}

extern "C" void kernel_launch(void* const* d_in, const int* in_sizes, int n_in,
                              void* d_out, int out_size, void* d_ws, size_t ws_size,
                              hipStream_t stream) {
    (void)in_sizes; (void)n_in; (void)d_ws; (void)ws_size; (void)out_size;
    const float* det = (const float*)d_in[0];       // (B, N, 5) float32
    float* out_final = (float*)d_out;               // (B, N) final scores
    float* out_keep  = out_final + NBATCH * NBOX;   // (B, N) keep mask as 0/1
    soft_nms_kernel<<<NBATCH, THREADS, 0, stream>>>(det, out_final, out_keep);
}